// PatchSelfAttention2_13469017440514
// MI455X (gfx1250) — compile-verified
//
#include <hip/hip_runtime.h>
#include <hip/hip_bf16.h>
#include <math.h>

typedef __attribute__((ext_vector_type(2))) float v2f;
typedef __attribute__((ext_vector_type(8))) float v8f;

#define EPS 1e-5f

__device__ __forceinline__ v8f wmma_f32(v2f a, v2f b, v8f c) {
    // D = A(16x4) * B(4x16) + C(16x16), fp32
    return __builtin_amdgcn_wmma_f32_16x16x4_f32(
        /*neg_a=*/false, a, /*neg_b=*/false, b,
        /*c_mod=*/(short)0, c, /*reuse_a=*/false, /*reuse_b=*/false);
}

// One wave computes a 16(M) x 64(N) strip of C = A[M,K] * B[K,N] (row-major),
// as 4 accumulators of 16x16. The A fragment is loaded once per k-step and
// reused across the 4 N-tiles (4x WMMA per load group).
__device__ __forceinline__ void gemm_strip_16x64(const float* __restrict__ Arow, int lda,
                                                 const float* __restrict__ Bcol, int ldb,
                                                 int K, v8f acc[4]) {
    const int lane = threadIdx.x & 31;
    const int half = lane >> 4;   // 0: lanes 0-15, 1: lanes 16-31
    const int l16  = lane & 15;
#pragma unroll
    for (int t = 0; t < 4; ++t) acc[t] = v8f{0.f, 0.f, 0.f, 0.f, 0.f, 0.f, 0.f, 0.f};

    const float* __restrict__ a_ptr = Arow + (size_t)l16 * lda + 2 * half;
    const float* __restrict__ b_ptr = Bcol + (size_t)(2 * half) * ldb + l16;
#pragma unroll 2
    for (int k = 0; k < K; k += 4) {
        v2f a;
        a.x = a_ptr[0];                       // A[l16][k + 2*half]
        a.y = a_ptr[1];                       // A[l16][k + 2*half + 1]
#pragma unroll
        for (int t = 0; t < 4; ++t) {
            v2f b;
            b.x = b_ptr[t * 16];              // B[k + 2*half    ][nB + t*16 + l16]
            b.y = b_ptr[t * 16 + (size_t)ldb];// B[k + 2*half + 1][nB + t*16 + l16]
            acc[t] = wmma_f32(a, b, acc[t]);
        }
        a_ptr += 4;
        b_ptr += (size_t)4 * ldb;
    }
}

// align_corners bilinear sample of a 32x32 channel at output pixel (y,x) of 128x128
__device__ __forceinline__ float bilin32(const float* __restrict__ ch, int y, int x) {
    const float sc = 31.0f / 127.0f;
    float fy = y * sc, fx = x * sc;
    int y0 = (int)fy, x0 = (int)fx;
    int y1 = min(y0 + 1, 31), x1 = min(x0 + 1, 31);
    float wy = fy - (float)y0, wx = fx - (float)x0;
    float v00 = ch[y0 * 32 + x0], v01 = ch[y0 * 32 + x1];
    float v10 = ch[y1 * 32 + x0], v11 = ch[y1 * 32 + x1];
    float top = v00 + (v01 - v00) * wx;
    float bot = v10 + (v11 - v10) * wx;
    return top + (bot - top) * wy;
}

__device__ __forceinline__ float sigmoidf(float x) { return 1.0f / (1.0f + expf(-x)); }

// ---------------------------------------------------------------------------
// Kernel 1: keysmall[n][128][1024] = bn_relu(conv1x1(other, [wk1;wk2]))
//   M=128 (rows 0-63: wk1/bnk1, 64-127: wk2/bnk2), K=512, N=1024, 8 batches
//   N-groups of 64: 1024/64 = 16. Waves = 8*8*16 = 1024.
// ---------------------------------------------------------------------------
__global__ void smallconv_kernel(const float* __restrict__ other,
                                 const float* __restrict__ wk1,
                                 const float* __restrict__ bnk1_s, const float* __restrict__ bnk1_b,
                                 const float* __restrict__ wk2,
                                 const float* __restrict__ bnk2_s, const float* __restrict__ bnk2_b,
                                 float* __restrict__ keysmall) {
    const int MT = 8, NG = 16;  // 128/16, 1024/64
    int wave = (blockIdx.x * blockDim.x + threadIdx.x) >> 5;
    if (wave >= 8 * MT * NG) return;  // whole-wave exit, EXEC stays uniform
    int n  = wave / (MT * NG);
    int r  = wave % (MT * NG);
    int mB = (r / NG) * 16, nB = (r % NG) * 64;

    const float* A = (mB < 64) ? (wk1 + (size_t)mB * 512) : (wk2 + (size_t)(mB - 64) * 512);
    const float* B = other + (size_t)n * 512 * 1024 + nB;
    v8f acc[4];
    gemm_strip_16x64(A, 512, B, 1024, 512, acc);

    const int lane = threadIdx.x & 31, half = lane >> 4, l16 = lane & 15;
    const float inv = rsqrtf(1.0f + EPS);
    float* O = keysmall + (size_t)n * 128 * 1024 + nB + l16;
#pragma unroll
    for (int i = 0; i < 8; ++i) {
        int m = mB + i + 8 * half;
        float sc, bi;
        if (m < 64) { sc = bnk1_s[m] * inv;      bi = bnk1_b[m]; }
        else        { sc = bnk2_s[m - 64] * inv; bi = bnk2_b[m - 64]; }
#pragma unroll
        for (int t = 0; t < 4; ++t)
            O[(size_t)m * 1024 + t * 16] = fmaxf(acc[t][i] * sc + bi, 0.0f);
    }
}

// ---------------------------------------------------------------------------
// Kernel 2: qv[n][192][16384] = conv1x1(branch0, [wv1;wq1;wq2]) with epilogue
//   rows 0-63: v (identity), 64-127: q1 (bn_relu bnq1), 128-191: q2 (bn_relu bnq2)
//   M=192, K=256, N=16384, 8 batches. N-groups: 16384/64 = 256. Waves = 24576.
// ---------------------------------------------------------------------------
__global__ void bigconv_kernel(const float* __restrict__ x,
                               const float* __restrict__ wv1,
                               const float* __restrict__ wq1,
                               const float* __restrict__ bnq1_s, const float* __restrict__ bnq1_b,
                               const float* __restrict__ wq2,
                               const float* __restrict__ bnq2_s, const float* __restrict__ bnq2_b,
                               float* __restrict__ qv) {
    const int MT = 12, NG = 256;  // 192/16, 16384/64
    int wave = (blockIdx.x * blockDim.x + threadIdx.x) >> 5;
    if (wave >= 8 * MT * NG) return;
    int n  = wave / (MT * NG);
    int r  = wave % (MT * NG);
    int mB = (r / NG) * 16, nB = (r % NG) * 64;

    const float* A;
    if (mB < 64)       A = wv1 + (size_t)mB * 256;
    else if (mB < 128) A = wq1 + (size_t)(mB - 64) * 256;
    else               A = wq2 + (size_t)(mB - 128) * 256;
    const float* B = x + (size_t)n * 256 * 16384 + nB;
    v8f acc[4];
    gemm_strip_16x64(A, 256, B, 16384, 256, acc);

    const int lane = threadIdx.x & 31, half = lane >> 4, l16 = lane & 15;
    const float inv = rsqrtf(1.0f + EPS);
    float* O = qv + (size_t)n * 192 * 16384 + nB + l16;
#pragma unroll
    for (int i = 0; i < 8; ++i) {
        int m = mB + i + 8 * half;
        bool bn = (m >= 64);
        float sc = 1.0f, bi = 0.0f;
        if (bn) {
            if (m < 128) { sc = bnq1_s[m - 64] * inv;  bi = bnq1_b[m - 64]; }
            else         { sc = bnq2_s[m - 128] * inv; bi = bnq2_b[m - 128]; }
        }
#pragma unroll
        for (int t = 0; t < 4; ++t) {
            float val = acc[t][i];
            if (bn) val = fmaxf(val * sc + bi, 0.0f);
            O[(size_t)m * 16384 + t * 16] = val;
        }
    }
}

// ---------------------------------------------------------------------------
// Kernel 3: patch gates. One block per (n, c_local) with c_local in [0,32).
//   sim[p][q] = sigmoid( sum_l qp[p,l] * kp[q,l] ), gate[p] = rowmean + colmean
//   k is bilinearly upsampled on the fly from keysmall (already bn_relu'd).
//   cOff = 0 for scale 1 (ph=32, P=16), cOff = 32 for scale 2 (ph=16, P=64).
// ---------------------------------------------------------------------------
__global__ void gate_kernel(const float* __restrict__ qv,
                            const float* __restrict__ keysmall,
                            int cOff, int ph, int gw, int P,
                            float* __restrict__ gateOut) {
    __shared__ float sim[64 * 64];  // 16 KB, covers both scales
    int n = blockIdx.x >> 5;
    int c = blockIdx.x & 31;
    const float* qch = qv + ((size_t)n * 192 + 64 + cOff + c) * 16384;       // q1 channel
    const float* kch = keysmall + ((size_t)n * 128 + cOff + c) * 1024;       // k1 channel (32x32)

    for (int pq = threadIdx.x; pq < P * P; pq += blockDim.x) {
        int p  = pq / P, qi = pq % P;
        int py0 = (p / gw) * ph,  px0 = (p % gw) * ph;
        int qy0 = (qi / gw) * ph, qx0 = (qi % gw) * ph;
        float dot = 0.0f;
        for (int py = 0; py < ph; ++py) {
            const float* qrow = qch + (size_t)(py0 + py) * 128 + px0;
            int ky = qy0 + py;
            for (int px = 0; px < ph; ++px) {
                dot += qrow[px] * bilin32(kch, ky, qx0 + px);
            }
        }
        sim[pq] = sigmoidf(dot);
    }
    __syncthreads();
    if ((int)threadIdx.x < P) {
        int p = threadIdx.x;
        float srow = 0.0f, scol = 0.0f;
        for (int qi = 0; qi < P; ++qi) {
            srow += sim[p * P + qi];
            scol += sim[qi * P + p];
        }
        gateOut[(size_t)blockIdx.x * P + p] = (srow + scol) / (float)P;
    }
}

// ---------------------------------------------------------------------------
// Kernel 4: build context[n][128][16384]
//   ch <  32: alpha * gate0[patch32] * v[ch]
//   ch <  64: alpha * gate2[patch16] * v[ch]
//   ch >= 64: gamma * v[ch-64] * sigmoid(q2[ch-64] * upsample(k2[ch-64]))
// ---------------------------------------------------------------------------
__global__ void context_kernel(const float* __restrict__ qv,
                               const float* __restrict__ keysmall,
                               const float* __restrict__ gate0,
                               const float* __restrict__ gate2,
                               const float* __restrict__ alpha,
                               const float* __restrict__ gamma,
                               float* __restrict__ ctx) {
    const size_t total = (size_t)8 * 128 * 16384;
    float a = alpha[0], g = gamma[0];
    for (size_t idx = (size_t)blockIdx.x * blockDim.x + threadIdx.x; idx < total;
         idx += (size_t)gridDim.x * blockDim.x) {
        int s  = (int)(idx & 16383);
        int ch = (int)((idx >> 14) & 127);
        int n  = (int)(idx >> 21);
        int y = s >> 7, x = s & 127;
        float out;
        if (ch < 64) {
            float vv = qv[((size_t)n * 192 + ch) * 16384 + s];
            float gate;
            if (ch < 32) {
                int p = (y >> 5) * 4 + (x >> 5);
                gate = gate0[((size_t)n * 32 + ch) * 16 + p];
            } else {
                int p = (y >> 4) * 8 + (x >> 4);
                gate = gate2[((size_t)n * 32 + (ch - 32)) * 64 + p];
            }
            out = a * gate * vv;
        } else {
            int c = ch - 64;
            float vv = qv[((size_t)n * 192 + c) * 16384 + s];
            float q2 = qv[((size_t)n * 192 + 128 + c) * 16384 + s];
            float k2 = bilin32(keysmall + ((size_t)n * 128 + 64 + c) * 1024, y, x);
            out = g * vv * sigmoidf(q2 * k2);
        }
        ctx[idx] = out;
    }
}

// ---------------------------------------------------------------------------
// Kernel 5: out[n][256][16384] = wz[256x128] * context[n][128][16384]
//   N-groups: 16384/64 = 256. Waves = 8*16*256 = 32768.
// ---------------------------------------------------------------------------
__global__ void outconv_kernel(const float* __restrict__ wz,
                               const float* __restrict__ ctx,
                               float* __restrict__ out) {
    const int MT = 16, NG = 256;  // 256/16, 16384/64
    int wave = (blockIdx.x * blockDim.x + threadIdx.x) >> 5;
    if (wave >= 8 * MT * NG) return;
    int n  = wave / (MT * NG);
    int r  = wave % (MT * NG);
    int mB = (r / NG) * 16, nB = (r % NG) * 64;

    const float* A = wz + (size_t)mB * 128;
    const float* B = ctx + (size_t)n * 128 * 16384 + nB;
    v8f acc[4];
    gemm_strip_16x64(A, 128, B, 16384, 128, acc);

    const int lane = threadIdx.x & 31, half = lane >> 4, l16 = lane & 15;
    float* O = out + (size_t)n * 256 * 16384 + nB + l16;
#pragma unroll
    for (int i = 0; i < 8; ++i) {
        int m = mB + i + 8 * half;
#pragma unroll
        for (int t = 0; t < 4; ++t)
            O[(size_t)m * 16384 + t * 16] = acc[t][i];
    }
}

// ---------------------------------------------------------------------------
extern "C" void kernel_launch(void* const* d_in, const int* in_sizes, int n_in,
                              void* d_out, int out_size, void* d_ws, size_t ws_size,
                              hipStream_t stream) {
    const float* branch0 = (const float*)d_in[0];
    const float* other   = (const float*)d_in[1];
    const float* wq1     = (const float*)d_in[2];
    const float* bnq1_s  = (const float*)d_in[3];
    const float* bnq1_b  = (const float*)d_in[4];
    const float* wk1     = (const float*)d_in[5];
    const float* bnk1_s  = (const float*)d_in[6];
    const float* bnk1_b  = (const float*)d_in[7];
    const float* wv1     = (const float*)d_in[8];
    const float* wq2     = (const float*)d_in[9];
    const float* bnq2_s  = (const float*)d_in[10];
    const float* bnq2_b  = (const float*)d_in[11];
    const float* wk2     = (const float*)d_in[12];
    const float* bnk2_s  = (const float*)d_in[13];
    const float* bnk2_b  = (const float*)d_in[14];
    const float* wz      = (const float*)d_in[15];
    const float* alpha   = (const float*)d_in[16];
    const float* gamma   = (const float*)d_in[17];
    float* out = (float*)d_out;

    // workspace layout (floats)
    float* ws = (float*)d_ws;
    float* keysmall = ws;                                  //  8*128*1024   =  1,048,576
    float* qv       = keysmall + (size_t)8 * 128 * 1024;   //  8*192*16384  = 25,165,824
    float* gate0    = qv + (size_t)8 * 192 * 16384;        //  8*32*16
    float* gate2    = gate0 + (size_t)8 * 32 * 16;         //  8*32*64
    float* ctx      = gate2 + (size_t)8 * 32 * 64;         //  8*128*16384  = 16,777,216

    const int BLK = 256;  // 8 waves per block

    // 1) k1/k2 small conv + bn_relu -> keysmall   (waves = 8*8*16 = 1024)
    smallconv_kernel<<<1024 / 8, BLK, 0, stream>>>(other, wk1, bnk1_s, bnk1_b,
                                                   wk2, bnk2_s, bnk2_b, keysmall);

    // 2) v/q1/q2 big conv -> qv                   (waves = 8*12*256 = 24576)
    bigconv_kernel<<<24576 / 8, BLK, 0, stream>>>(branch0, wv1,
                                                  wq1, bnq1_s, bnq1_b,
                                                  wq2, bnq2_s, bnq2_b, qv);

    // 3) patch gates, scale 1 (ph=32, gw=4, P=16) and scale 2 (ph=16, gw=8, P=64)
    gate_kernel<<<256, BLK, 0, stream>>>(qv, keysmall, /*cOff=*/0,  32, 4, 16, gate0);
    gate_kernel<<<256, BLK, 0, stream>>>(qv, keysmall, /*cOff=*/32, 16, 8, 64, gate2);

    // 4) build context
    context_kernel<<<65536, BLK, 0, stream>>>(qv, keysmall, gate0, gate2, alpha, gamma, ctx);

    // 5) final conv -> out                        (waves = 8*16*256 = 32768)
    outconv_kernel<<<32768 / 8, BLK, 0, stream>>>(wz, ctx, out);
}